// GCNNet_20083267076737
// MI455X (gfx1250) — compile-verified
//
#include <hip/hip_runtime.h>

typedef __attribute__((ext_vector_type(2))) float v2f;
typedef __attribute__((ext_vector_type(8))) float v8f;
typedef __attribute__((ext_vector_type(4))) unsigned int v4u;
typedef __attribute__((ext_vector_type(8))) int v8i;
typedef __attribute__((ext_vector_type(4))) int v4i;

#define N_NODES    50000
#define N_EDGES    800000
#define EMB        128
#define HID        256
#define OUTD       128
#define NUM_GRAPHS 256

#define KC     64      // K panel staged in LDS
#define NPANEL 64      // N panel per workgroup
#define MTILES (N_NODES / 16)   // 3125 exactly

#if __has_builtin(__builtin_amdgcn_tensor_load_to_lds)
#define HAVE_TDM 1
#endif

__device__ __forceinline__ void atomAddF(float* p, float v) {
  // relaxed agent-scope fp32 atomic -> global_atomic_add_f32 (no return)
  __hip_atomic_fetch_add(p, v, __ATOMIC_RELAXED, __HIP_MEMORY_SCOPE_AGENT);
}

#ifdef HAVE_TDM
// Stage a [rows x cols] fp32 tile (row stride = strideElems) from global into
// LDS at byte offset lds_off, via the Tensor Data Mover (ISA ch.8 D# layout).
__device__ __forceinline__ void tdm_stage(const float* gsrc, unsigned lds_off,
                                          int rows, int cols, int strideElems) {
  unsigned long long ga = (unsigned long long)gsrc;
  v4u g0;
  g0.x = 1u;                                             // count=1 (valid), user mode
  g0.y = lds_off;                                        // lds_addr (bytes)
  g0.z = (unsigned)(ga & 0xffffffffu);                   // global_addr[31:0]
  g0.w = (unsigned)((ga >> 32) & 0x1ffffffu) | (2u << 30); // global_addr[56:32] | type=2
  v8i g1;
  g1[0] = (2 << 16);                                     // wg_mask=0, data_size=4B
  g1[1] = (cols & 0xffff) << 16;                         // tensor_dim0[15:0]
  g1[2] = ((unsigned)cols >> 16) | ((rows & 0xffff) << 16); // tensor_dim0[31:16] | tensor_dim1[15:0]
  g1[3] = ((unsigned)rows >> 16) | ((cols & 0xffff) << 16); // tensor_dim1[31:16] | tile_dim0
  g1[4] = (rows & 0xffff);                               // tile_dim1 | tile_dim2=0 (2D)
  g1[5] = strideElems;                                   // tensor_dim0_stride[31:0]
  g1[6] = 0;                                             // stride[47:32] | dim1_stride[15:0]
  g1[7] = 0;
  v4i g2 = {0, 0, 0, 0};
  v4i g3 = {0, 0, 0, 0};
#if __clang_major__ >= 23
  v8i g4 = {0, 0, 0, 0, 0, 0, 0, 0};
  __builtin_amdgcn_tensor_load_to_lds(g0, g1, g2, g3, g4, 0);
#else
  __builtin_amdgcn_tensor_load_to_lds(g0, g1, g2, g3, 0);
#endif
}
#endif

__global__ __launch_bounds__(256) void k_zero(float* __restrict__ p, int n) {
  int i = blockIdx.x * blockDim.x + threadIdx.x;
  if (i < n) p[i] = 0.0f;
}

__global__ __launch_bounds__(256) void k_gather(const float* __restrict__ embed,
                                                const int* __restrict__ ids,
                                                float* __restrict__ x) {
  const int chunks = EMB / 4;
  int tid = blockIdx.x * blockDim.x + threadIdx.x;
  if (tid >= N_NODES * chunks) return;
  int i = tid / chunks;
  int c = (tid - i * chunks) << 2;
  *(float4*)(x + (size_t)i * EMB + c) =
      *(const float4*)(embed + (size_t)ids[i] * EMB + c);
}

__global__ __launch_bounds__(256) void k_deg(const int* __restrict__ dst,
                                             float* __restrict__ deg) {
  int e = blockIdx.x * blockDim.x + threadIdx.x;
  if (e < N_EDGES) atomAddF(&deg[dst[e]], 1.0f);
}

__global__ __launch_bounds__(256) void k_dinv(const float* __restrict__ deg,
                                              float* __restrict__ dinv) {
  int i = blockIdx.x * blockDim.x + threadIdx.x;
  if (i < N_NODES) dinv[i] = rsqrtf(deg[i] + 1.0f);
}

// H[M,N] = X[M,K] @ W[K,N] via fp32 WMMA 16x16x4.
// Block = 8 waves; each wave owns one 16-row M tile and a 64-wide N panel
// (4 accumulator tiles, A fragment reused 4x). W K-panels (64x64, 16 KB) are
// double-buffered in LDS, streamed by the Tensor Data Mover from wave 0.
// A layout (ISA 16x4 f32): lanes 0-15 -> K{0,1}, lanes 16-31 -> K{2,3}.
// C/D layout: VGPR v holds M=v (lanes 0-15) / M=v+8 (lanes 16-31), N=lane&15.
__global__ __launch_bounds__(256) void k_gemm_wmma(const float* __restrict__ X,
                                                   const float* __restrict__ W,
                                                   float* __restrict__ H,
                                                   int K, int N) {
  __shared__ float ldsW[2 * KC * NPANEL];   // 32 KB

  const int waveId = threadIdx.x >> 5;
  const int lane   = threadIdx.x & 31;
  const int lm     = lane & 15;
  const int hi     = lane >> 4;      // 0 or 1
  const int khalf  = hi << 1;        // K sub-offset 0 or 2

  const int  mTile = blockIdx.x * 8 + waveId;
  const bool valid = (mTile < MTILES);     // wave-uniform: EXEC stays all-1s
  const int  n0    = blockIdx.y * NPANEL;
  const int  P     = K / KC;

#ifdef HAVE_TDM
  unsigned ldsBase = (unsigned)(unsigned long long)(&ldsW[0]); // low 32b = LDS offset
  if (waveId == 0) {
    tdm_stage(W + n0, ldsBase, KC, NPANEL, N);
    __builtin_amdgcn_s_wait_tensorcnt(0);
  }
  __syncthreads();
#endif

  v8f acc0 = {}, acc1 = {}, acc2 = {}, acc3 = {};
  const float* xrow = X + (size_t)((mTile << 4) + lm) * K + khalf;

  for (int p = 0; p < P; ++p) {
    const float* lw = ldsW + (p & 1) * (KC * NPANEL);

#ifdef HAVE_TDM
    if (waveId == 0 && p + 1 < P)
      tdm_stage(W + (size_t)(p + 1) * KC * N + n0,
                ldsBase + (unsigned)(((p + 1) & 1) * (KC * NPANEL * 4)),
                KC, NPANEL, N);
#else
    __syncthreads();   // all reads of this buffer done
    {
      float* lwm = ldsW + (p & 1) * (KC * NPANEL);
      const float* gp = W + (size_t)p * KC * N + n0;
      for (int idx = threadIdx.x; idx < KC * (NPANEL / 4); idx += 256) {
        int r  = idx / (NPANEL / 4);
        int c4 = (idx - r * (NPANEL / 4)) << 2;
        *(float4*)(lwm + r * NPANEL + c4) = *(const float4*)(gp + (size_t)r * N + c4);
      }
    }
    __syncthreads();
#endif

    if (valid) {
#pragma unroll 4
      for (int k = 0; k < KC; k += 4) {
        v2f a = *(const v2f*)(xrow + p * KC + k);       // contiguous K,K+1
        const float* lr = lw + (k + khalf) * NPANEL + lm;
        v2f b0, b1, b2, b3;
        b0.x = lr[0];   b0.y = lr[NPANEL + 0];
        b1.x = lr[16];  b1.y = lr[NPANEL + 16];
        b2.x = lr[32];  b2.y = lr[NPANEL + 32];
        b3.x = lr[48];  b3.y = lr[NPANEL + 48];
        acc0 = __builtin_amdgcn_wmma_f32_16x16x4_f32(false, a, false, b0, (short)0, acc0, false, false);
        acc1 = __builtin_amdgcn_wmma_f32_16x16x4_f32(false, a, false, b1, (short)0, acc1, false, false);
        acc2 = __builtin_amdgcn_wmma_f32_16x16x4_f32(false, a, false, b2, (short)0, acc2, false, false);
        acc3 = __builtin_amdgcn_wmma_f32_16x16x4_f32(false, a, false, b3, (short)0, acc3, false, false);
      }
    }

#ifdef HAVE_TDM
    if (waveId == 0 && p + 1 < P) __builtin_amdgcn_s_wait_tensorcnt(0);
    __syncthreads();
#endif
  }

  if (valid) {
    float* hbase = H + (size_t)((mTile << 4) + (hi << 3)) * N + n0 + lm;
#pragma unroll
    for (int v = 0; v < 8; ++v) {
      hbase[(size_t)v * N + 0]  = acc0[v];
      hbase[(size_t)v * N + 16] = acc1[v];
      hbase[(size_t)v * N + 32] = acc2[v];
      hbase[(size_t)v * N + 48] = acc3[v];
    }
  }
}

__global__ __launch_bounds__(256) void k_edge_agg(const float* __restrict__ H,
                                                  float* __restrict__ AGG,
                                                  const int* __restrict__ src,
                                                  const int* __restrict__ dst,
                                                  const float* __restrict__ dinv,
                                                  int d) {
  int chunks = d >> 2;
  int tid = blockIdx.x * blockDim.x + threadIdx.x;
  if (tid >= N_EDGES * chunks) return;          // max 51.2M, fits int32
  int e = tid / chunks;
  int c = (tid - e * chunks) << 2;
  int s = src[e], t = dst[e];
  float nrm = dinv[s] * dinv[t];
  float4 h = *(const float4*)(H + (size_t)s * d + c);
  float* ap = AGG + (size_t)t * d + c;
  atomAddF(ap + 0, h.x * nrm);
  atomAddF(ap + 1, h.y * nrm);
  atomAddF(ap + 2, h.z * nrm);
  atomAddF(ap + 3, h.w * nrm);
}

__global__ __launch_bounds__(256) void k_finalize(float* __restrict__ AGG,
                                                  const float* __restrict__ H,
                                                  const float* __restrict__ dinv,
                                                  const float* __restrict__ bias,
                                                  const float* __restrict__ resid,
                                                  int d, int doRelu) {
  int chunks = d >> 2;
  int tid = blockIdx.x * blockDim.x + threadIdx.x;
  if (tid >= N_NODES * chunks) return;
  int i = tid / chunks;
  int c = (tid - i * chunks) << 2;
  float di = dinv[i];
  float d2 = di * di;
  float4 a = *(float4*)(AGG + (size_t)i * d + c);
  float4 h = *(const float4*)(H + (size_t)i * d + c);
  float4 b = *(const float4*)(bias + c);
  a.x += h.x * d2 + b.x;
  a.y += h.y * d2 + b.y;
  a.z += h.z * d2 + b.z;
  a.w += h.w * d2 + b.w;
  if (resid) {
    float4 r = *(const float4*)(resid + (size_t)i * d + c);
    a.x += r.x; a.y += r.y; a.z += r.z; a.w += r.w;
  }
  if (doRelu) {
    a.x = fmaxf(a.x, 0.0f); a.y = fmaxf(a.y, 0.0f);
    a.z = fmaxf(a.z, 0.0f); a.w = fmaxf(a.w, 0.0f);
  }
  *(float4*)(AGG + (size_t)i * d + c) = a;
}

__global__ __launch_bounds__(256) void k_cnt(const int* __restrict__ batch,
                                             float* __restrict__ cnt) {
  int i = blockIdx.x * blockDim.x + threadIdx.x;
  if (i < N_NODES) atomAddF(&cnt[batch[i]], 1.0f);
}

__global__ __launch_bounds__(256) void k_pool(const float* __restrict__ X,
                                              const int* __restrict__ batch,
                                              float* __restrict__ out) {
  const int chunks = OUTD / 4;
  int tid = blockIdx.x * blockDim.x + threadIdx.x;
  if (tid >= N_NODES * chunks) return;
  int i = tid / chunks;
  int c = (tid - i * chunks) << 2;
  int g = batch[i];
  float4 v = *(const float4*)(X + (size_t)i * OUTD + c);
  float* op = out + (size_t)g * OUTD + c;
  atomAddF(op + 0, v.x);
  atomAddF(op + 1, v.y);
  atomAddF(op + 2, v.z);
  atomAddF(op + 3, v.w);
}

__global__ __launch_bounds__(256) void k_div(float* __restrict__ out,
                                             const float* __restrict__ cnt) {
  int tid = blockIdx.x * blockDim.x + threadIdx.x;
  if (tid < NUM_GRAPHS * OUTD) out[tid] /= fmaxf(cnt[tid / OUTD], 1.0f);
}

static inline int nblocks(long long n) { return (int)((n + 255) / 256); }

extern "C" void kernel_launch(void* const* d_in, const int* in_sizes, int n_in,
                              void* d_out, int out_size, void* d_ws, size_t ws_size,
                              hipStream_t stream) {
  const int*   node_ids = (const int*)d_in[0];
  const int*   edge_idx = (const int*)d_in[1];
  const int*   batch    = (const int*)d_in[2];
  const float* embed    = (const float*)d_in[3];
  const float* W_in  = (const float*)d_in[4];
  const float* b_in  = (const float*)d_in[5];
  const float* W_h1  = (const float*)d_in[6];
  const float* b_h1  = (const float*)d_in[7];
  const float* W_h2  = (const float*)d_in[8];
  const float* b_h2  = (const float*)d_in[9];
  const float* W_out = (const float*)d_in[10];
  const float* b_out = (const float*)d_in[11];
  const int* src = edge_idx;
  const int* dst = edge_idx + N_EDGES;
  float* out = (float*)d_out;

  const size_t NODE_F = (size_t)N_NODES * HID;   // 12.8M floats per slab
  float* buf0 = (float*)d_ws;
  float* buf1 = buf0 + NODE_F;
  float* buf2 = buf1 + NODE_F;
  float* dinv = buf2 + NODE_F;
  float* deg  = dinv + N_NODES;
  float* cnt  = deg + N_NODES;

  const int B = 256;
  const dim3 gHID((MTILES + 7) / 8, HID / NPANEL);   // (391, 4)
  const dim3 gOUT((MTILES + 7) / 8, OUTD / NPANEL);  // (391, 2)

  // degrees -> dinv
  k_zero<<<nblocks(N_NODES), B, 0, stream>>>(deg, N_NODES);
  k_deg <<<nblocks(N_EDGES), B, 0, stream>>>(dst, deg);
  k_dinv<<<nblocks(N_NODES), B, 0, stream>>>(deg, dinv);

  // embedding gather -> buf0 [N, 128]
  k_gather<<<nblocks((long long)N_NODES * (EMB / 4)), B, 0, stream>>>(embed, node_ids, buf0);

  // ---- layer 1: 128 -> 256, no residual, ReLU ----
  k_gemm_wmma<<<gHID, B, 0, stream>>>(buf0, W_in, buf1, EMB, HID);
  k_zero<<<nblocks((long long)N_NODES * HID), B, 0, stream>>>(buf2, N_NODES * HID);
  k_edge_agg<<<nblocks((long long)N_EDGES * (HID / 4)), B, 0, stream>>>(buf1, buf2, src, dst, dinv, HID);
  k_finalize<<<nblocks((long long)N_NODES * (HID / 4)), B, 0, stream>>>(buf2, buf1, dinv, b_in, nullptr, HID, 1);
  // x = buf2

  // ---- layer 2: 256 -> 256, residual, ReLU ----
  k_gemm_wmma<<<gHID, B, 0, stream>>>(buf2, W_h1, buf0, HID, HID);
  k_zero<<<nblocks((long long)N_NODES * HID), B, 0, stream>>>(buf1, N_NODES * HID);
  k_edge_agg<<<nblocks((long long)N_EDGES * (HID / 4)), B, 0, stream>>>(buf0, buf1, src, dst, dinv, HID);
  k_finalize<<<nblocks((long long)N_NODES * (HID / 4)), B, 0, stream>>>(buf1, buf0, dinv, b_h1, buf2, HID, 1);
  // x = buf1

  // ---- layer 3: 256 -> 256, residual, ReLU ----
  k_gemm_wmma<<<gHID, B, 0, stream>>>(buf1, W_h2, buf2, HID, HID);
  k_zero<<<nblocks((long long)N_NODES * HID), B, 0, stream>>>(buf0, N_NODES * HID);
  k_edge_agg<<<nblocks((long long)N_EDGES * (HID / 4)), B, 0, stream>>>(buf2, buf0, src, dst, dinv, HID);
  k_finalize<<<nblocks((long long)N_NODES * (HID / 4)), B, 0, stream>>>(buf0, buf2, dinv, b_h2, buf1, HID, 1);
  // x = buf0

  // ---- layer 4: 256 -> 128, no residual, no ReLU ----
  k_gemm_wmma<<<gOUT, B, 0, stream>>>(buf0, W_out, buf1, HID, OUTD);
  k_zero<<<nblocks((long long)N_NODES * OUTD), B, 0, stream>>>(buf2, N_NODES * OUTD);
  k_edge_agg<<<nblocks((long long)N_EDGES * (OUTD / 4)), B, 0, stream>>>(buf1, buf2, src, dst, dinv, OUTD);
  k_finalize<<<nblocks((long long)N_NODES * (OUTD / 4)), B, 0, stream>>>(buf2, buf1, dinv, b_out, nullptr, OUTD, 0);
  // x = buf2 [N, 128]

  // ---- scatter-mean pool -> out [256, 128] ----
  k_zero<<<nblocks(NUM_GRAPHS), B, 0, stream>>>(cnt, NUM_GRAPHS);
  k_zero<<<nblocks(NUM_GRAPHS * OUTD), B, 0, stream>>>(out, NUM_GRAPHS * OUTD);
  k_cnt <<<nblocks(N_NODES), B, 0, stream>>>(batch, cnt);
  k_pool<<<nblocks((long long)N_NODES * (OUTD / 4)), B, 0, stream>>>(buf2, batch, out);
  k_div <<<nblocks(NUM_GRAPHS * OUTD), B, 0, stream>>>(out, cnt);
}